// DigitCaps_15814069583893
// MI455X (gfx1250) — compile-verified
//
#include <hip/hip_runtime.h>

typedef __attribute__((ext_vector_type(2))) float v2f;
typedef __attribute__((ext_vector_type(8))) float v8f;

#define IC      1152      // input capsules
#define WD      16        // out-capsule dim
#define KC      8         // in-capsule dim
#define BT      16        // batch tile (WMMA M)
#define HN      10        // out capsules
#define NB      256       // batch
#define WAVES   16
#define THREADS 512
#define IPW     (IC / WAVES)   // 72 capsules per wave

// LDS layout (floats)
#define L_OFF    0                       // routing logits l[i*16 + b]   (18432)
#define PART_OFF (IC * BT)               // per-wave s partials          (4096)
#define SV_OFF   (PART_OFF + WAVES*256)  // s / v tile [16b][16w]        (256)
#define RED_OFF  (SV_OFF + 256)          // reduction scratch            (512)
#define MX_OFF   (RED_OFF + 512)         // softmax max per batch        (16)
#define INV_OFF  (MX_OFF + 16)           // 1/denominator per batch      (16)
#define SMEM_FLOATS (INV_OFF + 16)       // 23328 floats = 93312 bytes

__global__ __launch_bounds__(THREADS)
void digitcaps_routing(const float* __restrict__ x,   // [256, 1152*8] flat
                       const float* __restrict__ W,   // [10, 1152, 16, 8] flat
                       float* __restrict__ out)       // [256, 10, 16]
{
  extern __shared__ float sm[];
  float* l   = sm + L_OFF;
  float* prt = sm + PART_OFF;
  float* sv  = sm + SV_OFF;
  float* red = sm + RED_OFF;
  float* mx  = sm + MX_OFF;
  float* inv = sm + INV_OFF;

  const int tid  = threadIdx.x;
  const int wave = tid >> 5;
  const int lane = tid & 31;
  const int row  = lane & 15;   // A: batch row / B: out-dim col / C: N index
  const int half = lane >> 4;   // selects K sub-pair / C upper 8 rows

  const int h  = blockIdx.x % HN;
  const int b0 = (blockIdx.x / HN) * BT;

  for (int idx = tid; idx < IC * BT; idx += THREADS) l[idx] = 0.0f;
  __syncthreads();

  const float* xrow = x + (size_t)(b0 + row) * (IC * KC); // per-lane batch row
  const float* wh   = W + (size_t)h * (IC * WD * KC);
  const int i0 = wave * IPW;

  for (int it = 0; it < 3; ++it) {
    // ---------- softmax over in-caps (axis=-2), with max subtraction ----------
    {
      const int bb = tid & 15, sl = tid >> 4;               // 32 slices x 16 batches
      float m = -3.402823466e38f;
      for (int i = sl; i < IC; i += 32) m = fmaxf(m, l[i * 16 + bb]);
      red[sl * 16 + bb] = m;
    }
    __syncthreads();
    if (tid < 16) {
      float m = -3.402823466e38f;
      for (int s = 0; s < 32; ++s) m = fmaxf(m, red[s * 16 + tid]);
      mx[tid] = m;
    }
    __syncthreads();
    {
      const int bb = tid & 15, sl = tid >> 4;
      float s = 0.0f, m = mx[bb];
      for (int i = sl; i < IC; i += 32) s += __expf(l[i * 16 + bb] - m);
      red[sl * 16 + bb] = s;
    }
    __syncthreads();
    if (tid < 16) {
      float s = 0.0f;
      for (int k = 0; k < 32; ++k) s += red[k * 16 + tid];
      inv[tid] = 1.0f / s;
    }
    __syncthreads();

    // ---------- s[b,w] = sum_i c[b,i] * (W_i x_i): fold c into A, WMMA K-accumulate
    v8f acc = {};
    const float mrow = mx[row], irow = inv[row];
    for (int i = i0; i < i0 + IPW; ++i) {
      const float c = __expf(l[i * 16 + row] - mrow) * irow;
      const float2 a0 = *(const float2*)(xrow + i * KC + 2 * half);       // k = 2h,2h+1
      const float2 a1 = *(const float2*)(xrow + i * KC + 4 + 2 * half);   // k = 4+2h,...
      const float2 w0 = *(const float2*)(wh + (size_t)(i * WD + row) * KC + 2 * half);
      const float2 w1 = *(const float2*)(wh + (size_t)(i * WD + row) * KC + 4 + 2 * half);
      v2f A0 = {a0.x * c, a0.y * c};
      v2f A1 = {a1.x * c, a1.y * c};
      v2f B0 = {w0.x, w0.y};
      v2f B1 = {w1.x, w1.y};
      acc = __builtin_amdgcn_wmma_f32_16x16x4_f32(false, A0, false, B0, (short)0, acc, false, false);
      acc = __builtin_amdgcn_wmma_f32_16x16x4_f32(false, A1, false, B1, (short)0, acc, false, false);
    }
#pragma unroll
    for (int j = 0; j < 8; ++j)
      prt[wave * 256 + (j + 8 * half) * 16 + row] = acc[j];
    __syncthreads();

    if (tid < 256) {
      float s = 0.0f;
#pragma unroll
      for (int w = 0; w < WAVES; ++w) s += prt[w * 256 + tid];
      sv[tid] = s;                                  // s tile [m=b][n=w]
    }
    __syncthreads();

    // ---------- squash along w ----------
    if (tid < 16) {
      float sq = 0.0f;
#pragma unroll
      for (int n = 0; n < 16; ++n) { float t = sv[tid * 16 + n]; sq += t * t; }
      red[tid] = (sq / (1.0f + sq)) * rsqrtf(sq);
    }
    __syncthreads();
    if (tid < 256) sv[tid] *= red[tid >> 4];        // sv now holds v
    __syncthreads();

    if (it == 2) break;   // 3rd agreement never affects output v

    // ---------- agreement[b,i] = sum_w u[b,i,w]*v[b,w]; b_logits += agreement ----
    float vv[8];
#pragma unroll
    for (int j = 0; j < 8; ++j) vv[j] = sv[(j + 8 * half) * 16 + row]; // v in C layout
    for (int i = i0; i < i0 + IPW; ++i) {
      v8f u = {};
      const float2 a0 = *(const float2*)(xrow + i * KC + 2 * half);
      const float2 a1 = *(const float2*)(xrow + i * KC + 4 + 2 * half);
      const float2 w0 = *(const float2*)(wh + (size_t)(i * WD + row) * KC + 2 * half);
      const float2 w1 = *(const float2*)(wh + (size_t)(i * WD + row) * KC + 4 + 2 * half);
      v2f A0 = {a0.x, a0.y};
      v2f A1 = {a1.x, a1.y};
      v2f B0 = {w0.x, w0.y};
      v2f B1 = {w1.x, w1.y};
      u = __builtin_amdgcn_wmma_f32_16x16x4_f32(false, A0, false, B0, (short)0, u, false, false);
      u = __builtin_amdgcn_wmma_f32_16x16x4_f32(false, A1, false, B1, (short)0, u, false, false);
      float r[8];
#pragma unroll
      for (int j = 0; j < 8; ++j) r[j] = u[j] * vv[j];
      // reduce over N (16 lanes within each half): xor masks < 16 stay in the half
#pragma unroll
      for (int j = 0; j < 8; ++j) {
        r[j] += __shfl_xor(r[j], 1);
        r[j] += __shfl_xor(r[j], 2);
        r[j] += __shfl_xor(r[j], 4);
        r[j] += __shfl_xor(r[j], 8);
      }
      if (row == 0) {
#pragma unroll
        for (int j = 0; j < 8; ++j)
          l[i * 16 + j + 8 * half] += r[j];
      }
    }
    __syncthreads();
  }

  // write v: out[b, h, w]
  if (tid < 256) {
    const int m = tid >> 4, n = tid & 15;
    out[(size_t)(b0 + m) * (HN * WD) + h * WD + n] = sv[tid];
  }
}

extern "C" void kernel_launch(void* const* d_in, const int* in_sizes, int n_in,
                              void* d_out, int out_size, void* d_ws, size_t ws_size,
                              hipStream_t stream) {
  (void)in_sizes; (void)n_in; (void)d_ws; (void)ws_size; (void)out_size;
  const float* x = (const float*)d_in[0];   // [256,32,8,6,6] -> flat [256, 9216]
  const float* W = (const float*)d_in[1];   // [1,10,1152,16,8]
  float* out = (float*)d_out;               // [256,10,16]

  const size_t smem = (size_t)SMEM_FLOATS * sizeof(float); // ~93 KB (<< 320 KB/WGP)
  (void)hipFuncSetAttribute(reinterpret_cast<const void*>(&digitcaps_routing),
                            hipFuncAttributeMaxDynamicSharedMemorySize, (int)smem);

  dim3 grid((NB / BT) * HN);   // 160 workgroups: 16 batch tiles x 10 digit caps
  dim3 block(THREADS);         // 16 wave32s
  digitcaps_routing<<<grid, block, smem, stream>>>(x, W, out);
}